// PCQActivation_54915451846700
// MI455X (gfx1250) — compile-verified
//
#include <hip/hip_runtime.h>
#include <math.h>

typedef float v4f __attribute__((ext_vector_type(4)));

#define SMOOTH 0.999f
#define QMAX   255.0f
#define DCOLS  4096
#define TPB    256
#define MAXC   64   // padded cluster capacity in workspace

// ---- workspace layout -------------------------------------------------------
// floats: wsf[0..MAXC)        cluster running min   (init +inf each call)
//         wsf[MAXC..2*MAXC)   cluster running max   (init -inf each call)
//         wsf[2*MAXC..3*MAXC) per-cluster scale s
//         wsf[3*MAXC..4*MAXC) per-cluster zero  z
// ints:   wsi[0..C]           segment row-start prefix (seg_start[C] == N)
//         wsi[MAXC+1 .. ]     segment cluster id
// -----------------------------------------------------------------------------

__device__ __forceinline__ float hswish(float x) {
    return x * fminf(fmaxf(x + 3.0f, 0.0f), 6.0f) * (1.0f / 6.0f);
}

// sign-aware float atomic min/max via integer ordering (correct for mixed signs)
__device__ __forceinline__ void atomicMinF(float* addr, float v) {
    if (v >= 0.0f) atomicMin((int*)addr, __float_as_int(v));
    else           atomicMax((unsigned int*)addr, __float_as_uint(v));
}
__device__ __forceinline__ void atomicMaxF(float* addr, float v) {
    if (v >= 0.0f) atomicMax((int*)addr, __float_as_int(v));
    else           atomicMin((unsigned int*)addr, __float_as_uint(v));
}

// K0: reset cluster min/max, build segment prefix + id tables. One tiny block.
__global__ void k_init(const int* __restrict__ cinfo, int C,
                       float* __restrict__ wsf, int* __restrict__ wsi) {
    const int t = threadIdx.x;
    if (t < C) {
        wsf[t]        = INFINITY;   // cluster min
        wsf[MAXC + t] = -INFINITY;  // cluster max
    }
    if (t == 0) {
        int acc = 0;
#pragma unroll 1
        for (int j = 0; j < C; ++j) {
            wsi[j]            = acc;            // seg_start
            wsi[MAXC + 1 + j] = cinfo[2 * j];   // cluster id
            acc += cinfo[2 * j + 1];            // count
        }
        wsi[C] = acc;                           // == N
    }
}

// K1: one block per row. The row is staged into LDS with CDNA5 async DMA
// (global_load_async_to_lds_b128, tracked by ASYNCcnt), consumed with staged
// s_wait_asynccnt waits. Async ops complete in order per wave, and each lane
// reads back only the LDS slot its own async op wrote, so no block barrier is
// needed. Then hardswish + row min/max -> atomic per-cluster min/max.
__global__ void __launch_bounds__(TPB)
k_rowreduce(const float* __restrict__ x, int C,
            float* __restrict__ wsf, const int* __restrict__ wsi) {
    constexpr int CH = DCOLS / (4 * TPB);      // 4 chunks of 4KB per block
    const int row = blockIdx.x;
    const int t   = threadIdx.x;
    const v4f* rp4 = (const v4f*)(x + (size_t)row * DCOLS);

    __shared__ v4f buf[CH][TPB];               // 16KB: whole row staged in LDS

    // Issue all async DMA loads up front (4 in flight per wave).
#pragma unroll
    for (int i = 0; i < CH; ++i) {
        const unsigned lds = (unsigned)(size_t)&buf[i][t];
        const unsigned long long ga = (unsigned long long)(&rp4[i * TPB + t]);
        asm volatile("global_load_async_to_lds_b128 %0, %1, off"
                     :: "v"(lds), "v"(ga) : "memory");
    }

    float mn = INFINITY, mx = -INFINITY;
#pragma unroll
    for (int i = 0; i < CH; ++i) {
        // chunk i complete once at most (CH-1-i) async ops remain outstanding
        if      (CH - 1 - i == 3) asm volatile("s_wait_asynccnt 0x3" ::: "memory");
        else if (CH - 1 - i == 2) asm volatile("s_wait_asynccnt 0x2" ::: "memory");
        else if (CH - 1 - i == 1) asm volatile("s_wait_asynccnt 0x1" ::: "memory");
        else                      asm volatile("s_wait_asynccnt 0x0" ::: "memory");
        v4f v = buf[i][t];
#pragma unroll
        for (int k = 0; k < 4; ++k) {
            float h = hswish(v[k]);
            mn = fminf(mn, h);
            mx = fmaxf(mx, h);
        }
    }

    // wave32 butterfly reduction
#pragma unroll
    for (int off = 16; off >= 1; off >>= 1) {
        mn = fminf(mn, __shfl_xor(mn, off, 32));
        mx = fmaxf(mx, __shfl_xor(mx, off, 32));
    }
    __shared__ float smn[TPB / 32], smx[TPB / 32];
    const int wave = t >> 5, lane = t & 31;
    if (lane == 0) { smn[wave] = mn; smx[wave] = mx; }
    __syncthreads();
    if (t == 0) {
#pragma unroll
        for (int w = 1; w < TPB / 32; ++w) {
            mn = fminf(mn, smn[w]);
            mx = fmaxf(mx, smx[w]);
        }
        int seg = C - 1;                       // contiguous-segment lookup (C=10)
        for (int j = 0; j < C - 1; ++j) {
            if (row < wsi[j + 1]) { seg = j; break; }
        }
        const int id = wsi[MAXC + 1 + seg];
        atomicMinF(&wsf[id],        mn);
        atomicMaxF(&wsf[MAXC + id], mx);
    }
}

// K2: EMA range update + qparams; writes new_range into tail of d_out.
__global__ void k_params(const float* __restrict__ act_range, int C,
                         float* __restrict__ wsf, float* __restrict__ out_range) {
    const int c = threadIdx.x;
    if (c >= C) return;
    const float cm = wsf[c];
    const float cM = wsf[MAXC + c];
    const float lo = act_range[2 * c]     * SMOOTH + cm * (1.0f - SMOOTH);
    const float hi = act_range[2 * c + 1] * SMOOTH + cM * (1.0f - SMOOTH);
    out_range[2 * c]     = lo;
    out_range[2 * c + 1] = hi;
    const float s = (hi - lo) / QMAX;
    const float z = -rintf(lo / s);
    wsf[2 * MAXC + c] = s;
    wsf[3 * MAXC + c] = z;
}

// K3: recompute hardswish, fake-quantize, stream out with NT stores so the
// 268MB output stream does not evict x from the 192MB L2 (x was pulled in by
// pass 1 and is re-read here).
__global__ void __launch_bounds__(TPB)
k_quant(const float* __restrict__ x, int C,
        const float* __restrict__ wsf, const int* __restrict__ wsi,
        float* __restrict__ out) {
    const int row = blockIdx.x;
    const int t   = threadIdx.x;

    int seg = C - 1;                           // uniform per block -> scalar path
    for (int j = 0; j < C - 1; ++j) {
        if (row < wsi[j + 1]) { seg = j; break; }
    }
    const int   id    = wsi[MAXC + 1 + seg];
    const float s     = wsf[2 * MAXC + id];
    const float z     = wsf[3 * MAXC + id];
    const float inv_s = 1.0f / s;

    const v4f* rp4 = (const v4f*)(x   + (size_t)row * DCOLS);
    v4f*       op4 = (v4f*)      (out + (size_t)row * DCOLS);

#pragma unroll
    for (int i = 0; i < DCOLS / (4 * TPB); ++i) {
        v4f v = __builtin_nontemporal_load(&rp4[i * TPB + t]);  // last use of x
        v4f o;
#pragma unroll
        for (int k = 0; k < 4; ++k) {
            const float h = hswish(v[k]);
            const float q = fminf(fmaxf(rintf(h * inv_s + z), 0.0f), QMAX);
            o[k] = (q - z) * s;
        }
        __builtin_nontemporal_store(o, &op4[i * TPB + t]);      // NT: keep L2 for x
    }
}

extern "C" void kernel_launch(void* const* d_in, const int* in_sizes, int n_in,
                              void* d_out, int out_size, void* d_ws, size_t ws_size,
                              hipStream_t stream) {
    const float* x         = (const float*)d_in[0];
    const float* act_range = (const float*)d_in[1];
    const int*   cinfo     = (const int*)d_in[2];

    const int ND = in_sizes[0];          // N * D
    const int C  = in_sizes[1] / 2;      // clusters
    const int N  = ND / DCOLS;

    float* out       = (float*)d_out;
    float* out_range = out + (size_t)ND; // tuple: out_q then new_range

    float* wsf = (float*)d_ws;
    int*   wsi = (int*)(wsf + 4 * MAXC);

    k_init     <<<1, 64,  0, stream>>>(cinfo, C, wsf, wsi);
    k_rowreduce<<<N, TPB, 0, stream>>>(x, C, wsf, wsi);
    k_params   <<<1, 64,  0, stream>>>(act_range, C, wsf, out_range);
    k_quant    <<<N, TPB, 0, stream>>>(x, C, wsf, wsi, out);
}